// Voxurf_41944650613250
// MI455X (gfx1250) — compile-verified
//
#include <hip/hip_runtime.h>
#include <math.h>

// Voxurf Alphas2Weights (segmented transmittance scan) for MI455X / gfx1250.
// Memory-bound (~200 MB traffic -> ~9 us at 23.3 TB/s). Strategy:
//   - one wave32 per ray (ray_id is sorted; avg 128 samples/ray)
//   - wave-level inclusive scan of log2(1-alpha) via cross-lane shuffles
//   - log2/exp2 space == reference ln/exp space (same clamp), maps to
//     v_log_f32 / v_exp_f32 directly
//   - boundary-detection tile staged into LDS via the Tensor Data Mover
//     (tensor_load_to_lds + s_wait_tensorcnt) when the toolchain exposes it
//   - global_prefetch_b8 streaming hints, coalesced loads

#define EPSV 1e-10f

// ------------------------------------------------------------- TDM plumbing
#if __has_builtin(__builtin_amdgcn_tensor_load_to_lds)
#define HAVE_TDM 1
#if __has_include(<hip/amd_detail/amd_gfx1250_TDM.h>)
#define TDM_6ARG 1   // amdgpu-toolchain (clang-23 / therock-10.0): 6-arg form
#endif
#endif

#ifdef HAVE_TDM
typedef unsigned int u32x4 __attribute__((ext_vector_type(4)));
typedef int          i32x4 __attribute__((ext_vector_type(4)));
typedef int          i32x8 __attribute__((ext_vector_type(8)));

// Issue one 1-D TDM transfer: tile_elems x 4-byte elements from gaddr into
// LDS at byte offset lds_off. tensor_dim0 = elems_avail so the tail tile is
// zero-filled by TDM OOB handling. All inputs are wave-uniform -> SGPR D#.
__device__ __forceinline__ void tdm_load_1d_b32(const void* gaddr,
                                                unsigned lds_off,
                                                int elems_avail,
                                                int tile_elems) {
    unsigned long long ga = (unsigned long long)(uintptr_t)gaddr;
    unsigned td0 = (unsigned)(elems_avail > 0 ? elems_avail : 0);

    u32x4 g0 = {0u, 0u, 0u, 0u};
    g0[0] = 1u;                                   // count=1, user descriptor
    g0[1] = lds_off;                              // lds_addr (bytes)
    g0[2] = (unsigned)(ga & 0xFFFFFFFFull);       // global_addr[31:0]
    g0[3] = (unsigned)((ga >> 32) & 0x01FFFFFFull)// global_addr[56:32]
            | (2u << 30);                         // type=2 ("image")

    i32x8 g1 = {0, 0, 0, 0, 0, 0, 0, 0};
    g1[0] = (int)(2u << 16);                      // data_size=2 -> 4 bytes
    g1[1] = (int)((td0 & 0xFFFFu) << 16);         // tensor_dim0[15:0]
    g1[2] = (int)((td0 >> 16) | (1u << 16));      // tensor_dim0[31:16] | tensor_dim1=1
    g1[3] = (int)(((unsigned)tile_elems & 0xFFFFu) << 16); // tile_dim0
    g1[4] = 0;                                    // tile_dim1=0 (1-D), tile_dim2=0
    g1[5] = (int)td0;                             // dim0 stride (unused for 1-D)

    i32x4 gz = {0, 0, 0, 0};
#ifdef TDM_6ARG
    i32x8 gz8 = {0, 0, 0, 0, 0, 0, 0, 0};
    __builtin_amdgcn_tensor_load_to_lds(g0, g1, gz, gz, gz8, 0);
#else
    __builtin_amdgcn_tensor_load_to_lds(g0, g1, gz, gz, 0);
#endif
}

__device__ __forceinline__ void tdm_wait0() {
#if __has_builtin(__builtin_amdgcn_s_wait_tensorcnt)
    __builtin_amdgcn_s_wait_tensorcnt(0);
#endif
}
#endif // HAVE_TDM

// ---------------------------------------------------------------- kernel 1
__global__ void k_init_starts(int* __restrict__ ray_start, int n_plus1, int M) {
    int r = blockIdx.x * blockDim.x + threadIdx.x;
    if (r < n_plus1) ray_start[r] = M;   // default: empty trailing rays
}

// ---------------------------------------------------------------- kernel 2
// ray_id is sorted; for each boundary i (ray_id[i] != ray_id[i-1]) set
// ray_start[r] = i for all r in (prev, cur].  Empty rays between boundaries
// get the same start as the next non-empty ray -> count 0.
__global__ __launch_bounds__(256)
void k_boundaries(const int* __restrict__ ray_id, int M,
                  int* __restrict__ ray_start) {
    __shared__ int tile[256];
    const int t         = threadIdx.x;
    const int blockBase = blockIdx.x * 256;

#ifdef HAVE_TDM
    // Each wave DMAs its own 32-int sub-tile into LDS via the TDM.
    const int wave  = t >> 5;
    const int wbase = blockBase + wave * 32;
    if (wbase < M) {
        unsigned lds_off = (unsigned)(uintptr_t)(&tile[0])
                         + (unsigned)(wave * 32 * 4);
        tdm_load_1d_b32(ray_id + wbase, lds_off, M - wbase, 32);
    }
    tdm_wait0();          // wave waits for its own descriptor
    __syncthreads();      // publish tile across waves
#else
    {
        int i0 = blockBase + t;
        tile[t] = (i0 < M) ? ray_id[i0] : 0;
        __syncthreads();
    }
#endif

    const int i = blockBase + t;
    if (i >= M) return;
    int cur  = tile[t];
    int prev = (t == 0) ? ((i == 0) ? -1 : ray_id[i - 1]) : tile[t - 1];
    if (cur != prev) {
        for (int r = prev + 1; r <= cur; ++r) ray_start[r] = i;
    }
}

// ---------------------------------------------------------------- kernel 3
// One wave32 per ray. Sequential chunks of 32 samples; per chunk an
// inclusive wave scan of log2(1-alpha) gives per-sample exclusive
// log-transmittance; carry chains chunks.
__global__ __launch_bounds__(256)
void k_scan(const float* __restrict__ alpha,
            const float* __restrict__ rgb,
            const int*   __restrict__ ray_start,
            int n_rays,
            float* __restrict__ weights,
            float* __restrict__ alphainv_last,
            float* __restrict__ rgb_out) {
    const int lane = threadIdx.x & 31;
    const int wave = threadIdx.x >> 5;
    const int r = blockIdx.x * (blockDim.x >> 5) + wave;
    if (r >= n_rays) return;                 // wave-uniform: whole wave exits

    const int start = ray_start[r];
    const int end   = ray_start[r + 1];

    float carry = 0.0f;                      // running sum of log2(1-alpha)
    float cr = 0.0f, cg = 0.0f, cb = 0.0f;   // weighted rgb accumulators

    for (int base = start; base < end; base += 32) {
        const int  i   = base + lane;
        const bool act = (i < end);

        // stream ~4 chunks ahead (speculative prefetch: OOB silently dropped)
        if (base + 128 < end) {
            __builtin_prefetch(&alpha[base + 128 + lane], 0, 1);
            __builtin_prefetch(&rgb[(size_t)(base + 128 + lane) * 3], 0, 1);
        }

        float a = act ? alpha[i] : 0.0f;               // inactive lanes: lg=0
        float om = fminf(fmaxf(1.0f - a, EPSV), 1.0f); // clip(1-a, EPS, 1)
        float lg = __log2f(om);                        // v_log_f32

        // wave32 inclusive scan of lg
        float inc = lg;
        #pragma unroll
        for (int off = 1; off < 32; off <<= 1) {
            float t = __shfl_up(inc, off, 32);
            if (lane >= off) inc += t;
        }
        float excl = inc - lg;                 // exclusive within chunk
        float logT = carry + excl;             // exclusive within ray
        float w = exp2f(logT) * a;             // v_exp_f32

        if (act) {
            weights[i] = w;
            size_t j = (size_t)i * 3;
            cr = fmaf(w, rgb[j + 0], cr);
            cg = fmaf(w, rgb[j + 1], cg);
            cb = fmaf(w, rgb[j + 2], cb);
        }
        carry += __shfl(inc, 31, 32);          // chunk total -> carry
    }

    // wave reduction of rgb accumulators
    #pragma unroll
    for (int off = 16; off > 0; off >>= 1) {
        cr += __shfl_down(cr, off, 32);
        cg += __shfl_down(cg, off, 32);
        cb += __shfl_down(cb, off, 32);
    }

    if (lane == 0) {
        alphainv_last[r] = (end > start) ? exp2f(carry) : 1.0f;
        size_t j = (size_t)r * 3;
        rgb_out[j + 0] = cr;
        rgb_out[j + 1] = cg;
        rgb_out[j + 2] = cb;
    }
}

// ---------------------------------------------------------------- launcher
extern "C" void kernel_launch(void* const* d_in, const int* in_sizes, int n_in,
                              void* d_out, int out_size, void* d_ws, size_t ws_size,
                              hipStream_t stream) {
    const float* alpha  = (const float*)d_in[0];
    const float* rgb    = (const float*)d_in[1];
    const int*   ray_id = (const int*)d_in[2];   // harness: integer -> int32

    const int M = in_sizes[0];
    // out = weights[M] ++ alphainv_last[N] ++ rgb_marched[N*3]  => N derived:
    const int N = (out_size - M) / 4;

    float* weights       = (float*)d_out;
    float* alphainv_last = weights + M;
    float* rgb_out       = alphainv_last + N;

    int* ray_start = (int*)d_ws;                 // (N+1) ints of scratch

    {   // ray_start[0..N] = M
        int n1 = N + 1;
        k_init_starts<<<(n1 + 255) / 256, 256, 0, stream>>>(ray_start, n1, M);
    }
    {   // fill segment starts from sorted ray_id (TDM-staged LDS tile)
        k_boundaries<<<(M + 255) / 256, 256, 0, stream>>>(ray_id, M, ray_start);
    }
    {   // one wave32 per ray, 8 waves per 256-thread block
        const int waves_per_block = 256 / 32;
        int blocks = (N + waves_per_block - 1) / waves_per_block;
        k_scan<<<blocks, 256, 0, stream>>>(alpha, rgb, ray_start, N,
                                           weights, alphainv_last, rgb_out);
    }
}